// preprocessor_52132313038907
// MI455X (gfx1250) — compile-verified
//
#include <hip/hip_runtime.h>

// Problem constants (reference: N=50000, E=800000, D=128)
#define DIM   128
#define LDA   132   // padded LDS row stride (words); 132%64==4 -> conflict-free b64 frag reads

typedef __attribute__((ext_vector_type(2))) float v2f;
typedef __attribute__((ext_vector_type(8))) float v8f;

// Guaranteed-hardware float atomic add (no CAS-loop expansion risk).
// global_atomic_add_f32 without return -> tracked by STOREcnt; kernel-end
// S_ENDPGM performs an implicit wait-idle, so no explicit s_wait needed.
__device__ __forceinline__ void atomic_add_f32_hw(float* p, float v) {
  asm volatile("global_atomic_add_f32 %0, %1, off" : : "v"(p), "v"(v) : "memory");
}

// ---------------------------------------------------------------------------
// Kernel 1: support = x @ W_gc + b_gc
// Block = 256 threads (8 waves), computes a 16-row x 128-col strip.
// Wave w computes the 16x16 tile at columns [16w, 16w+16).
// f32 WMMA 16x16x4: A-frag lane layout: row = lane&15, v0 = K + 2*(lane>>4),
// v1 = K + 2*(lane>>4) + 1. B-frag mirrors with row<->col swapped.
// C layout: vgpr v holds M = v + 8*(lane>>4), N = lane&15.
// ---------------------------------------------------------------------------
__global__ __launch_bounds__(256)
void gemm_bias_kernel(const float* __restrict__ X, const float* __restrict__ W,
                      const float* __restrict__ bias, float* __restrict__ Y) {
  __shared__ float sA[16 * LDA];
  const int m0  = blockIdx.x * 16;
  const int tid = threadIdx.x;

  // Stage 16x128 A-strip (contiguous in global) into padded LDS
  const float4* xsrc = (const float4*)(X + (size_t)m0 * DIM);
#pragma unroll
  for (int i = 0; i < 2; ++i) {
    int idx = tid + 256 * i;                 // float4 index within strip
    int r = (idx * 4) >> 7, c = (idx * 4) & 127;
    *(float4*)&sA[r * LDA + c] = xsrc[idx];
  }
  __syncthreads();

  const int lane = tid & 31;
  const int half = lane >> 4;                // 0: lanes 0-15, 1: lanes 16-31
  const int l    = lane & 15;
  const int n0   = (tid >> 5) * 16;          // wave's N-tile origin

  v8f c = {};
#pragma unroll 4
  for (int k = 0; k < DIM; k += 4) {
    v2f a, b;
    const float* ap = &sA[l * LDA + k + 2 * half];
    a.x = ap[0];
    a.y = ap[1];
    b.x = W[(size_t)(k + 2 * half) * DIM + n0 + l];
    b.y = W[(size_t)(k + 2 * half + 1) * DIM + n0 + l];
    c = __builtin_amdgcn_wmma_f32_16x16x4_f32(false, a, false, b, (short)0, c,
                                              false, false);
  }

  const float bv = bias[n0 + l];
#pragma unroll
  for (int v = 0; v < 8; ++v) {
    int row = m0 + v + 8 * half;
    Y[(size_t)row * DIM + n0 + l] = c[v] + bv;
  }
}

// ---------------------------------------------------------------------------
// Kernel 2: agg[dst[e]] += support[src[e]] * vals[e]   (scatter SpMM)
// One wave per edge; each lane handles 4 contiguous features (float4 gather,
// 4 hardware float atomic adds). support/agg are 25.6 MB each -> resident in
// the 192MB L2, so this pass is L2 gather/atomic bound, not HBM bound.
// ---------------------------------------------------------------------------
__global__ __launch_bounds__(256)
void spmm_scatter_kernel(const float* __restrict__ support,
                         const float* __restrict__ vals,
                         const int* __restrict__ src,
                         const int* __restrict__ dst,
                         float* __restrict__ agg, int nE) {
  int e = blockIdx.x * 8 + (threadIdx.x >> 5);
  if (e >= nE) return;
  int lane = threadIdx.x & 31;
  int s = src[e];
  int d = dst[e];
  float v = vals[e];
  float4 m = ((const float4*)(support + (size_t)s * DIM))[lane];
  float* drow = agg + (size_t)d * DIM + lane * 4;
  atomic_add_f32_hw(drow + 0, m.x * v);
  atomic_add_f32_hw(drow + 1, m.y * v);
  atomic_add_f32_hw(drow + 2, m.z * v);
  atomic_add_f32_hw(drow + 3, m.w * v);
}

// ---------------------------------------------------------------------------
// Kernel 3: x2 = relu(agg) @ W2^T + b2 ; out = x2 / ||x2||_row
// Same WMMA tiling as kernel 1; B accessed transposed: B[k][n] = W2[n*128+k]
// (contiguous float2 per lane). Results staged in LDS for the row-norm
// reduction, then normalized and stored coalesced.
// ---------------------------------------------------------------------------
__global__ __launch_bounds__(256)
void gemm2_relu_norm_kernel(const float* __restrict__ AGG,
                            const float* __restrict__ W2,
                            const float* __restrict__ B2,
                            float* __restrict__ OUT) {
  __shared__ float sA[16 * LDA];
  __shared__ float sO[16 * LDA];
  __shared__ float sP[16 * 16];
  __shared__ float sNrm[16];

  const int m0  = blockIdx.x * 16;
  const int tid = threadIdx.x;

  // Stage relu(agg) strip into LDS
  const float4* asrc = (const float4*)(AGG + (size_t)m0 * DIM);
#pragma unroll
  for (int i = 0; i < 2; ++i) {
    int idx = tid + 256 * i;
    int r = (idx * 4) >> 7, c = (idx * 4) & 127;
    float4 f = asrc[idx];
    f.x = fmaxf(f.x, 0.0f);
    f.y = fmaxf(f.y, 0.0f);
    f.z = fmaxf(f.z, 0.0f);
    f.w = fmaxf(f.w, 0.0f);
    *(float4*)&sA[r * LDA + c] = f;
  }
  __syncthreads();

  const int lane = tid & 31;
  const int half = lane >> 4;
  const int l    = lane & 15;
  const int n0   = (tid >> 5) * 16;

  v8f c = {};
#pragma unroll 4
  for (int k = 0; k < DIM; k += 4) {
    v2f a, b;
    const float* ap = &sA[l * LDA + k + 2 * half];
    a.x = ap[0];
    a.y = ap[1];
    // B = W2^T : element (k, n) = W2[n*DIM + k] -> contiguous pair per lane
    const float* bp = &W2[(size_t)(n0 + l) * DIM + k + 2 * half];
    b.x = bp[0];
    b.y = bp[1];
    c = __builtin_amdgcn_wmma_f32_16x16x4_f32(false, a, false, b, (short)0, c,
                                              false, false);
  }

  const float bv = B2[n0 + l];
#pragma unroll
  for (int v = 0; v < 8; ++v) {
    int row = v + 8 * half;
    sO[row * LDA + n0 + l] = c[v] + bv;
  }
  __syncthreads();

  // Row sum-of-squares: 16 threads per row, 8 elements each, two-stage reduce
  {
    int r = tid >> 4, q = tid & 15;
    float p = 0.0f;
#pragma unroll
    for (int j = 0; j < 8; ++j) {
      float t = sO[r * LDA + q * 8 + j];
      p += t * t;
    }
    sP[r * 16 + q] = p;
  }
  __syncthreads();
  if (tid < 16) {
    float s = 0.0f;
#pragma unroll
    for (int j = 0; j < 16; ++j) s += sP[tid * 16 + j];
    sNrm[tid] = 1.0f / sqrtf(s);
  }
  __syncthreads();

  // Normalized, coalesced store
#pragma unroll
  for (int i = 0; i < 8; ++i) {
    int idx = tid + 256 * i;
    int r = idx >> 7, cc = idx & 127;
    OUT[(size_t)(m0 + r) * DIM + cc] = sO[r * LDA + cc] * sNrm[r];
  }
}

// ---------------------------------------------------------------------------
extern "C" void kernel_launch(void* const* d_in, const int* in_sizes, int n_in,
                              void* d_out, int out_size, void* d_ws, size_t ws_size,
                              hipStream_t stream) {
  const float* x    = (const float*)d_in[0];
  const float* vals = (const float*)d_in[1];
  const float* Wgc  = (const float*)d_in[2];
  const float* bgc  = (const float*)d_in[3];
  const float* W2   = (const float*)d_in[4];
  const float* b2   = (const float*)d_in[5];
  const int*   src  = (const int*)d_in[6];
  const int*   dst  = (const int*)d_in[7];

  const int N = in_sizes[0] / DIM;   // 50000
  const int E = in_sizes[1];         // 800000

  float* support = (float*)d_ws;                     // N*DIM floats
  float* agg     = support + (size_t)N * DIM;        // N*DIM floats

  // agg must be zero before scatter-add (graph-capture-safe async memset)
  hipMemsetAsync(agg, 0, (size_t)N * DIM * sizeof(float), stream);

  const int tilesM = N / 16;         // 3125 (exact)
  gemm_bias_kernel<<<tilesM, 256, 0, stream>>>(x, Wgc, bgc, support);
  spmm_scatter_kernel<<<(E + 7) / 8, 256, 0, stream>>>(support, vals, src, dst,
                                                       agg, E);
  gemm2_relu_norm_kernel<<<tilesM, 256, 0, stream>>>(agg, W2, b2, (float*)d_out);
}